// HybridNeuralODEModel_44659069944428
// MI455X (gfx1250) — compile-verified
//
#include <hip/hip_runtime.h>
#include <hip/hip_bf16.h>
#include <math.h>

#if !defined(__HIP_DEVICE_COMPILE__) || __has_builtin(__builtin_amdgcn_wmma_f32_16x16x4_f32)
#else
#error "missing __builtin_amdgcn_wmma_f32_16x16x4_f32 on this target"
#endif

typedef __attribute__((ext_vector_type(2))) float v2f;
typedef __attribute__((ext_vector_type(4))) float v4f;
typedef __attribute__((ext_vector_type(8))) float v8f;

#define WMMA_F32(a, b, c) \
  __builtin_amdgcn_wmma_f32_16x16x4_f32(false, (a), false, (b), (short)0, (c), false, false)

static constexpr int BATCH = 4096;
static constexpr int TLEN  = 256;
static constexpr int DIN   = 32;
static constexpr int PDIM  = 64;
static constexpr int HDIM  = 64;
static constexpr int QDIM  = 160;   // 5 * 32
static constexpr int NSTEPS = 10;
static constexpr float DT = 0.1f;
static constexpr float CLAMP_V = 10.0f;

__constant__ float DECLO[8] = {
  -0.010597401784997278f, 0.032883011666982945f, 0.030841381835986965f,
  -0.18703481171888114f, -0.02798376941698385f, 0.6308807679295904f,
   0.7148465705525415f,   0.23037781330885523f };
__constant__ float DECHI[8] = {
  -0.23037781330885523f, 0.7148465705525415f, -0.6308807679295904f,
  -0.02798376941698385f, 0.18703481171888114f, 0.030841381835986965f,
   0.032883011666982945f, -0.010597401784997278f };

__device__ __forceinline__ v8f vsplat(float s) {
  return (v8f){s, s, s, s, s, s, s, s};
}

// ---------------------------------------------------------------------------
// Kernel 1: setup. Wavelet adjoint functionals c[5][256], spectral norms of
// W1/W2 (power iteration), W1n/W2n, fused projection matrix M (64x160), biasH.
// ---------------------------------------------------------------------------
__global__ __launch_bounds__(256) void setup_kernel(
    const float* __restrict__ Wp, const float* __restrict__ bp,
    const float* __restrict__ Wf, const float* __restrict__ bf,
    const float* __restrict__ W1, const float* __restrict__ W2,
    float* __restrict__ c_out,  // 5*256
    float* __restrict__ S_out,  // 5
    float* __restrict__ W1n, float* __restrict__ W2n,
    float* __restrict__ M,      // 64*160, row-major [h][q]
    float* __restrict__ biasH)  // 64
{
  __shared__ float bufA[5][132];
  __shared__ float bufB[5][132];
  __shared__ float sv[64], yv[64], zv[64];
  __shared__ float snrm;
  const int tid = threadIdx.x;

  // ---- wavelet mean functionals via adjoint DWT (zero-pad db4, L=8) ----
  // forward level: a[m] = sum_j F[j] * x[2m+1-j];  lengths 256->131->69->38
  if (tid < 5) {
    float* cOut = c_out + tid * TLEN;
    if (tid == 4) {                      // time mean
      for (int t = 0; t < TLEN; ++t) cOut[t] = 1.0f / (float)TLEN;
    } else {
      int lens[4]; int filts[3]; int depth;
      if (tid == 0) { depth = 3; lens[0]=38; lens[1]=69; lens[2]=131; lens[3]=256;
                      filts[0]=0; filts[1]=0; filts[2]=0; }      // low: LO,LO,LO
      else if (tid == 1) { depth = 1; lens[0]=131; lens[1]=256; filts[0]=1; } // d1
      else if (tid == 2) { depth = 2; lens[0]=69; lens[1]=131; lens[2]=256;
                           filts[0]=1; filts[1]=0; }             // d2: HI then LO
      else { depth = 3; lens[0]=38; lens[1]=69; lens[2]=131; lens[3]=256;
             filts[0]=1; filts[1]=0; filts[2]=0; }               // d3: HI,LO,LO
      float* cur = bufA[tid];
      float* nxt = bufB[tid];
      const float inv0 = 1.0f / (float)lens[0];
      for (int m = 0; m < lens[0]; ++m) cur[m] = inv0;
      for (int li = 0; li < depth; ++li) {
        const int Tout = lens[li], Tin = lens[li + 1];
        const float* F = filts[li] ? DECHI : DECLO;
        float* dst = (li == depth - 1) ? cOut : nxt;
        for (int t = 0; t < Tin; ++t) {
          float s = 0.0f;
          for (int j = 0; j < 8; ++j) {
            const int idx = t + j - 1;           // 2m = t + j - 1
            if (idx >= 0 && ((idx & 1) == 0)) {
              const int m = idx >> 1;
              if (m < Tout) s += F[j] * cur[m];
            }
          }
          dst[t] = s;
        }
        nxt = cur;
        cur = dst;
      }
    }
  }
  __syncthreads();
  if (tid < 5) {
    float s = 0.0f;
    const float* cc = c_out + tid * TLEN;
    for (int t = 0; t < TLEN; ++t) s += cc[t];
    S_out[tid] = s;
  }
  __syncthreads();

  // ---- spectral norms (power iteration on W^T W), normalized weights ----
  for (int w = 0; w < 2; ++w) {
    const float* W = (w == 0) ? W1 : W2;
    float* Wn = (w == 0) ? W1n : W2n;
    if (tid < 64) sv[tid] = 1.0f;
    __syncthreads();
    for (int it = 0; it < 64; ++it) {
      if (tid < 64) {
        float s = 0.0f;
        for (int j = 0; j < 64; ++j) s += W[tid * 64 + j] * sv[j];
        yv[tid] = s;
      }
      __syncthreads();
      if (tid < 64) {
        float s = 0.0f;
        for (int i = 0; i < 64; ++i) s += W[i * 64 + tid] * yv[i];
        zv[tid] = s;
      }
      __syncthreads();
      if (tid == 0) {
        float s = 0.0f;
        for (int i = 0; i < 64; ++i) s += zv[i] * zv[i];
        snrm = rsqrtf(fmaxf(s, 1e-30f));
      }
      __syncthreads();
      if (tid < 64) sv[tid] = zv[tid] * snrm;
      __syncthreads();
    }
    if (tid == 0) {
      float s2 = 0.0f;
      for (int i = 0; i < 64; ++i) {
        float s = 0.0f;
        for (int j = 0; j < 64; ++j) s += W[i * 64 + j] * sv[j];
        s2 += s * s;
      }
      snrm = 1.0f / fmaxf(sqrtf(s2), 1e-20f);
    }
    __syncthreads();
    const float inv_sig = snrm;
    for (int e = tid; e < 64 * 64; e += 256) Wn[e] = W[e] * inv_sig;
    __syncthreads();
  }

  // ---- M[h, k*32+d] = sum_p Wf[h, k*64+p] * Wp[p, d] ----
  for (int idx = tid; idx < HDIM * QDIM; idx += 256) {
    const int h = idx / QDIM, q = idx % QDIM;
    const int k = q >> 5, d = q & 31;
    float s = 0.0f;
    for (int p = 0; p < PDIM; ++p)
      s += Wf[h * (5 * PDIM) + k * PDIM + p] * Wp[p * DIN + d];
    M[idx] = s;
  }
  // ---- biasH[h] = bf[h] + sum_k S_k * (Wf[h, k*P:...] . bp) ----
  if (tid < HDIM) {
    float bb = bf[tid];
    for (int k = 0; k < 5; ++k) {
      float t2 = 0.0f;
      for (int p = 0; p < PDIM; ++p)
        t2 += Wf[tid * (5 * PDIM) + k * PDIM + p] * bp[p];
      bb += S_out[k] * t2;
    }
    biasH[tid] = bb;
  }
}

// ---------------------------------------------------------------------------
// Kernel 2: Y[b, k*32+d] = sum_t c[k][t] * x[b, t, d]
// Bandwidth pass over x (128 MB): per-lane b128 loads; each wave pulls 512
// contiguous bytes (4 time steps x 32 channels) per vmem instruction.
// Thread map: d4 = tid&7 (channel quad), g = tid>>3 (time phase, 32 phases).
// ---------------------------------------------------------------------------
__global__ __launch_bounds__(256) void reduce_kernel(
    const float* __restrict__ x, const float* __restrict__ c,
    float* __restrict__ Y)
{
  __shared__ float sc[5 * TLEN];
  __shared__ __align__(16) float part[32 * 5 * 8 * 4];   // [g][k][d4][4] = 20KB
  const int tid = threadIdx.x;
  for (int i = tid; i < 5 * TLEN; i += 256) sc[i] = c[i];
  __syncthreads();

  const int b = blockIdx.x;
  const int d4 = tid & 7;     // channel quad: channels 4*d4 .. 4*d4+3
  const int g  = tid >> 3;    // time phase 0..31
  const float* xb = x + (size_t)b * (TLEN * DIN) + 4 * d4;

  v4f acc[5];
#pragma unroll
  for (int k = 0; k < 5; ++k) acc[k] = (v4f){0.f, 0.f, 0.f, 0.f};

  for (int t = g; t < TLEN; t += 32) {
    const v4f v = *(const v4f*)(xb + t * DIN);     // global_load_b128
#pragma unroll
    for (int k = 0; k < 5; ++k) acc[k] += sc[k * TLEN + t] * v;
  }

  v4f* pp = (v4f*)part;
#pragma unroll
  for (int k = 0; k < 5; ++k) pp[g * 40 + k * 8 + d4] = acc[k];
  __syncthreads();

  if (tid < QDIM) {
    const int k = tid >> 5, d = tid & 31;
    float s = 0.0f;
    for (int gg = 0; gg < 32; ++gg)
      s += part[(gg * 40 + k * 8 + (d >> 2)) * 4 + (d & 3)];
    Y[(size_t)b * QDIM + tid] = s;
  }
}

// ---------------------------------------------------------------------------
// Kernel 3: h0 GEMM + RK4 ODE (WMMA f32 16x16x4) + scalar readout, per 16-row tile.
// WMMA f32 layouts (wave32):
//   A (16x4):  lane L, vgpr j  <-> A[L%16, j + 2*(L/16)]
//   B (4x16):  lane L, vgpr j  <-> B[j + 2*(L/16), L%16]
//   C/D(16x16):lane L, vgpr j  <-> D[j + 8*(L/16), L%16]
// ---------------------------------------------------------------------------
__device__ __forceinline__ void stage_tile(float* st, const v8f x[4],
                                           int lm, int half) {
#pragma unroll
  for (int nt = 0; nt < 4; ++nt)
#pragma unroll
    for (int j = 0; j < 8; ++j)
      st[(j + 8 * half) * 64 + nt * 16 + lm] = x[nt][j];
}

// out[r, n] = bias[n] + sum_k in_tile[r, k] * W[n*64 + k]   (i.e. in @ W^T)
__device__ __forceinline__ void gemm16x64(const float* st, const float* W,
                                          const float* bias, v8f acc[4],
                                          int lm, int half) {
  const int koff = 2 * half;
#pragma unroll
  for (int nt = 0; nt < 4; ++nt) acc[nt] = vsplat(bias[nt * 16 + lm]);
#pragma unroll
  for (int c = 0; c < 16; ++c) {
    v2f a = *(const v2f*)(st + lm * 64 + 4 * c + koff);
#pragma unroll
    for (int nt = 0; nt < 4; ++nt) {
      v2f b = *(const v2f*)(W + (nt * 16 + lm) * 64 + 4 * c + koff);
      acc[nt] = WMMA_F32(a, b, acc[nt]);
    }
  }
}

// dh = clamp( silu(x @ W1n^T + b1) @ W2n^T + b2 , +-10 )
__device__ __forceinline__ void feval(float* st, const float* sW1,
                                      const float* sW2, const float* sb1,
                                      const float* sb2, const v8f x[4],
                                      v8f dh[4], int lm, int half) {
  v8f u[4];
  stage_tile(st, x, lm, half);           // per-wave LDS; DS ops in-order
  gemm16x64(st, sW1, sb1, u, lm, half);
#pragma unroll
  for (int nt = 0; nt < 4; ++nt)
#pragma unroll
    for (int j = 0; j < 8; ++j) {
      const float v = u[nt][j];
      u[nt][j] = v / (1.0f + expf(-v));  // silu
    }
  stage_tile(st, u, lm, half);
  gemm16x64(st, sW2, sb2, dh, lm, half);
#pragma unroll
  for (int nt = 0; nt < 4; ++nt)
#pragma unroll
    for (int j = 0; j < 8; ++j)
      dh[nt][j] = fminf(CLAMP_V, fmaxf(-CLAMP_V, dh[nt][j]));
}

__global__ __launch_bounds__(64) void ode_kernel(
    const float* __restrict__ Y, const float* __restrict__ M,
    const float* __restrict__ biasH, const float* __restrict__ W1n,
    const float* __restrict__ W2n, const float* __restrict__ b1,
    const float* __restrict__ b2, const float* __restrict__ Wr1,
    const float* __restrict__ br1, const float* __restrict__ Wr2,
    const float* __restrict__ br2, float* __restrict__ out)
{
  __shared__ __align__(16) float sW1[64 * 64];
  __shared__ __align__(16) float sW2[64 * 64];
  __shared__ __align__(16) float stage[2][16 * 64];
  __shared__ float sb1[64], sb2[64];

  const int tid = threadIdx.x;
  const int warp = tid >> 5;
  const int lane = tid & 31;
  const int lm = lane & 15;
  const int half = lane >> 4;
  float* st = stage[warp];

  for (int e = tid; e < 64 * 64; e += 64) { sW1[e] = W1n[e]; sW2[e] = W2n[e]; }
  if (tid < 64) { sb1[tid] = b1[tid]; sb2[tid] = b2[tid]; }
  __syncthreads();

  const int row0 = blockIdx.x * 32 + warp * 16;

  // ---- h0 = Y(16x160) @ M^T + biasH, K=160 in 40 chunks of 4 ----
  v8f h[4];
  {
    const int koff = 2 * half;
#pragma unroll
    for (int nt = 0; nt < 4; ++nt) h[nt] = vsplat(biasH[nt * 16 + lm]);
    for (int c = 0; c < 40; ++c) {
      v2f a = *(const v2f*)(Y + (size_t)(row0 + lm) * QDIM + 4 * c + koff);
#pragma unroll
      for (int nt = 0; nt < 4; ++nt) {
        v2f b = *(const v2f*)(M + (size_t)(nt * 16 + lm) * QDIM + 4 * c + koff);
        h[nt] = WMMA_F32(a, b, h[nt]);
      }
    }
  }

  // ---- RK4 (3/8 rule), 10 fixed steps, per-row independent ----
  v8f k1[4], k2[4], k3[4], k4[4], tmp[4];
#pragma unroll 1
  for (int step = 0; step < NSTEPS; ++step) {
    feval(st, sW1, sW2, sb1, sb2, h, k1, lm, half);
#pragma unroll
    for (int nt = 0; nt < 4; ++nt)
      tmp[nt] = h[nt] + vsplat(DT / 3.0f) * k1[nt];
    feval(st, sW1, sW2, sb1, sb2, tmp, k2, lm, half);
#pragma unroll
    for (int nt = 0; nt < 4; ++nt)
      tmp[nt] = h[nt] + vsplat(DT) * k2[nt] - vsplat(DT / 3.0f) * k1[nt];
    feval(st, sW1, sW2, sb1, sb2, tmp, k3, lm, half);
#pragma unroll
    for (int nt = 0; nt < 4; ++nt)
      tmp[nt] = h[nt] + vsplat(DT) * (k1[nt] - k2[nt] + k3[nt]);
    feval(st, sW1, sW2, sb1, sb2, tmp, k4, lm, half);
#pragma unroll
    for (int nt = 0; nt < 4; ++nt)
      h[nt] = h[nt] + vsplat(DT / 8.0f) *
                          (k1[nt] + vsplat(3.0f) * k2[nt] +
                           vsplat(3.0f) * k3[nt] + k4[nt]);
  }

  // ---- readout: out = relu(h @ Wr1^T + br1) @ Wr2^T + br2 ----
  stage_tile(st, h, lm, half);
  __syncthreads();
  if (lane < 16) {
    float o = br2[0];
    for (int cc = 0; cc < 32; ++cc) {
      float s = br1[cc];
      for (int i = 0; i < 64; ++i) s += st[lane * 64 + i] * Wr1[cc * 64 + i];
      o += fmaxf(s, 0.0f) * Wr2[cc];
    }
    out[row0 + lane] = o;
  }
}

// ---------------------------------------------------------------------------
extern "C" void kernel_launch(void* const* d_in, const int* in_sizes, int n_in,
                              void* d_out, int out_size, void* d_ws,
                              size_t ws_size, hipStream_t stream) {
  (void)in_sizes; (void)n_in; (void)out_size; (void)ws_size;
  const float* x   = (const float*)d_in[0];
  const float* Wp  = (const float*)d_in[1];
  const float* bp  = (const float*)d_in[2];
  const float* Wf  = (const float*)d_in[3];
  const float* bf  = (const float*)d_in[4];
  const float* W1  = (const float*)d_in[5];
  const float* b1  = (const float*)d_in[6];
  const float* W2  = (const float*)d_in[7];
  const float* b2  = (const float*)d_in[8];
  const float* Wr1 = (const float*)d_in[9];
  const float* br1 = (const float*)d_in[10];
  const float* Wr2 = (const float*)d_in[11];
  const float* br2 = (const float*)d_in[12];
  float* out = (float*)d_out;
  float* ws  = (float*)d_ws;

  // workspace layout (floats), total ~2.6 MB
  float* c    = ws + 0;       // 5*256
  float* S    = ws + 1280;    // 5
  float* W1n  = ws + 1536;    // 4096
  float* W2n  = ws + 5632;    // 4096
  float* M    = ws + 9728;    // 64*160
  float* bH   = ws + 19968;   // 64
  float* Y    = ws + 20480;   // 4096*160

  setup_kernel<<<1, 256, 0, stream>>>(Wp, bp, Wf, bf, W1, W2,
                                      c, S, W1n, W2n, M, bH);
  reduce_kernel<<<BATCH, 256, 0, stream>>>(x, c, Y);
  ode_kernel<<<BATCH / 32, 64, 0, stream>>>(Y, M, bH, W1n, W2n, b1, b2,
                                            Wr1, br1, Wr2, br2, out);
}